// RelationalMultiAggrMP_45157286150353
// MI455X (gfx1250) — compile-verified
//
#include <hip/hip_runtime.h>
#include <math.h>

#define N_NODES   50000
#define HIDDEN    128
#define MSGW      64
#define NCOL      192      // 3*MSG
#define KDIM      256      // 2*HIDDEN
#define N_ETYPES  4
#define EPT       100000
#define N_EDGES   (N_ETYPES*EPT)
#define DELTA_C   1.1515f
#define SMALL_C   1e-7f
#define LDS_STRIDE 260     // 256 + 4 pad -> bank-conflict-free wave32 A reads
#define M_TILE    32       // edges per block (2 WMMA M-tiles per wave)

typedef float v2f __attribute__((ext_vector_type(2)));
typedef float v8f __attribute__((ext_vector_type(8)));

// ---------------- CSR build ----------------
__global__ void k_zero(int* __restrict__ p, int n) {
  int i = blockIdx.x * 256 + threadIdx.x;
  if (i < n) p[i] = 0;
}

__device__ __forceinline__ const long long* pick_adj(int t, const long long* a0,
    const long long* a1, const long long* a2, const long long* a3) {
  return (t == 0) ? a0 : (t == 1) ? a1 : (t == 2) ? a2 : a3;
}

__global__ void k_count(const long long* __restrict__ a0, const long long* __restrict__ a1,
                        const long long* __restrict__ a2, const long long* __restrict__ a3,
                        int* __restrict__ cnt) {
  int id = blockIdx.x * 256 + threadIdx.x;
  if (id >= N_EDGES) return;
  int t = id / EPT, i = id - t * EPT;
  const long long* a = pick_adj(t, a0, a1, a2, a3);
  int tgt = (int)a[2 * (size_t)i + 1];
  atomicAdd(&cnt[tgt], 1);
}

__global__ void k_scan(const int* __restrict__ cnt, int* __restrict__ rs, int n) {
  __shared__ int s[1024];
  __shared__ int running;
  int tid = threadIdx.x;
  if (tid == 0) running = 0;
  __syncthreads();
  for (int base = 0; base < n; base += 1024) {
    int i = base + tid;
    int v = (i < n) ? cnt[i] : 0;
    s[tid] = v;
    __syncthreads();
    for (int off = 1; off < 1024; off <<= 1) {
      int t = (tid >= off) ? s[tid - off] : 0;
      __syncthreads();
      s[tid] += t;
      __syncthreads();
    }
    if (i < n) rs[i] = running + (s[tid] - v);   // exclusive prefix
    __syncthreads();
    if (tid == 0) running += s[1023];
    __syncthreads();
  }
  if (tid == 0) rs[n] = running;
}

__global__ void k_fill(const long long* __restrict__ a0, const long long* __restrict__ a1,
                       const long long* __restrict__ a2, const long long* __restrict__ a3,
                       const int* __restrict__ rs, int* __restrict__ cursor,
                       int* __restrict__ el) {
  int id = blockIdx.x * 256 + threadIdx.x;
  if (id >= N_EDGES) return;
  int t = id / EPT, i = id - t * EPT;
  const long long* a = pick_adj(t, a0, a1, a2, a3);
  int tgt = (int)a[2 * (size_t)i + 1];
  int pos = atomicAdd(&cursor[tgt], 1);
  el[rs[tgt] + pos] = id;
}

// ---------------- WMMA edge GEMM:  msgs = relu([x[src],x[tgt]] @ W[e] + b[e]) --------
// Block = 128 threads (4 waves), 32 edges per block. Each wave: 2 M-tiles x 3 N-tiles,
// so 3 B-fragments (6 global loads) feed 6 v_wmma_f32_16x16x4_f32 per K-step.
__global__ __launch_bounds__(128)
void k_gemm(const float* __restrict__ x,
            const long long* __restrict__ a0, const long long* __restrict__ a1,
            const long long* __restrict__ a2, const long long* __restrict__ a3,
            const float* __restrict__ W, const float* __restrict__ bias,
            float* __restrict__ msgs) {
  const int et = blockIdx.y;
  const long long* adj = pick_adj(et, a0, a1, a2, a3);
  const float* We = W + (size_t)et * KDIM * NCOL;
  const float* be = bias + et * NCOL;
  const int m0 = blockIdx.x * M_TILE;   // 100000 = 3125*32 exact
  const int tid = threadIdx.x;

  __shared__ float As[M_TILE * LDS_STRIDE];
  __shared__ int rows[2 * M_TILE];      // [2e]=src, [2e+1]=tgt
  if (tid < 2 * M_TILE)
    rows[tid] = (int)adj[(size_t)(m0 + (tid >> 1)) * 2 + (tid & 1)];
  __syncthreads();

  // Gather 32 edges x 256 f32 into LDS as float4 (2048 float4, 16 per thread).
  for (int idx = tid; idx < 2048; idx += 128) {
    int e = idx >> 6, q = idx & 63;
    int half = q >> 5, f4 = q & 31;
    int r = rows[e * 2 + half];
    float4 v = ((const float4*)(x + (size_t)r * HIDDEN))[f4];
    *(float4*)&As[e * LDS_STRIDE + half * HIDDEN + f4 * 4] = v;
  }
  __syncthreads();

  const int lane  = tid & 31;
  const int wave  = tid >> 5;
  const int rowA  = lane & 15;             // A: M = lane%16
  const int khalf = (lane >> 4) << 1;      // A/B: K base offset 0 or 2
  const int nloc  = lane & 15;             // B/C: N = lane%16
  const int n0 = (wave + 0) * 16 + nloc;
  const int n1 = (wave + 4) * 16 + nloc;
  const int n2 = (wave + 8) * 16 + nloc;

  v8f acc00 = {}, acc01 = {}, acc02 = {};  // M-tile 0 (edges m0..m0+15)
  v8f acc10 = {}, acc11 = {}, acc12 = {};  // M-tile 1 (edges m0+16..m0+31)
  #pragma unroll 2
  for (int k = 0; k < KDIM; k += 4) {
    int kb = k + khalf;
    v2f aL = *(const v2f*)&As[rowA * LDS_STRIDE + kb];          // ds_load_b64
    v2f aH = *(const v2f*)&As[(rowA + 16) * LDS_STRIDE + kb];   // ds_load_b64
    const float* w0 = We + (size_t)kb * NCOL;
    const float* w1 = w0 + NCOL;
    v2f b0; b0.x = w0[n0]; b0.y = w1[n0];
    v2f b1; b1.x = w0[n1]; b1.y = w1[n1];
    v2f b2; b2.x = w0[n2]; b2.y = w1[n2];
    acc00 = __builtin_amdgcn_wmma_f32_16x16x4_f32(false, aL, false, b0, (short)0, acc00, false, false);
    acc10 = __builtin_amdgcn_wmma_f32_16x16x4_f32(false, aH, false, b0, (short)0, acc10, false, false);
    acc01 = __builtin_amdgcn_wmma_f32_16x16x4_f32(false, aL, false, b1, (short)0, acc01, false, false);
    acc11 = __builtin_amdgcn_wmma_f32_16x16x4_f32(false, aH, false, b1, (short)0, acc11, false, false);
    acc02 = __builtin_amdgcn_wmma_f32_16x16x4_f32(false, aL, false, b2, (short)0, acc02, false, false);
    acc12 = __builtin_amdgcn_wmma_f32_16x16x4_f32(false, aH, false, b2, (short)0, acc12, false, false);
  }

  // Epilogue: C layout -> lane half selects M=r or M=r+8. Bias + ReLU, store msgs rows.
  const int mhi = (lane >> 4) << 3;
  const long long ebase = (long long)et * EPT + m0;
  float bb0 = be[n0], bb1 = be[n1], bb2 = be[n2];
  #pragma unroll
  for (int r = 0; r < 8; r++) {
    size_t ro0 = (size_t)(ebase + mhi + r) * NCOL;        // M-tile 0
    size_t ro1 = (size_t)(ebase + 16 + mhi + r) * NCOL;   // M-tile 1
    float o;
    o = acc00[r] + bb0; msgs[ro0 + n0] = o > 0.f ? o : 0.f;
    o = acc01[r] + bb1; msgs[ro0 + n1] = o > 0.f ? o : 0.f;
    o = acc02[r] + bb2; msgs[ro0 + n2] = o > 0.f ? o : 0.f;
    o = acc10[r] + bb0; msgs[ro1 + n0] = o > 0.f ? o : 0.f;
    o = acc11[r] + bb1; msgs[ro1 + n1] = o > 0.f ? o : 0.f;
    o = acc12[r] + bb2; msgs[ro1 + n2] = o > 0.f ? o : 0.f;
  }
}

// ---------------- Per-node aggregation: one wave32 per node, 2 cols/lane ----------
__global__ __launch_bounds__(256)
void k_aggr(const float* __restrict__ msgs, const int* __restrict__ rs,
            const int* __restrict__ el, float* __restrict__ out) {
  int node = blockIdx.x * 8 + (threadIdx.x >> 5);
  int lane = threadIdx.x & 31;
  if (node >= N_NODES) return;
  int beg = rs[node], end = rs[node + 1];
  int deg = end - beg;

  float s0 = 0.f, s1 = 0.f, m0 = 0.f, m1 = 0.f;
  float x0 = -__builtin_inff(), x1 = -__builtin_inff();
  for (int j = beg; j < end; j++) {
    const float* row = msgs + (size_t)el[j] * NCOL;
    s0 += row[lane];          s1 += row[32 + lane];
    m0 += row[64 + lane];     m1 += row[96 + lane];
    x0 = fmaxf(x0, row[128 + lane]);
    x1 = fmaxf(x1, row[160 + lane]);
  }
  float denom = deg > 0 ? (float)deg : 1.0f;
  float mean0 = m0 / denom, mean1 = m1 / denom;

  float v0 = 0.f, v1 = 0.f;                    // second pass for variance
  for (int j = beg; j < end; j++) {
    const float* row = msgs + (size_t)el[j] * NCOL;
    float a = row[64 + lane], c = row[96 + lane];
    v0 += fmaxf(a * a - mean0 * mean0, 0.f) + SMALL_C;
    v1 += fmaxf(c * c - mean1 * mean1, 0.f) + SMALL_C;
  }
  bool he = deg > 0;
  float std0 = he ? sqrtf(v0) : 0.f;
  float std1 = he ? sqrtf(v1) : 0.f;
  float mx0 = he ? x0 : 0.f, mx1 = he ? x1 : 0.f;

  float ld  = logf((float)deg + 1.0f);
  float amp = ld / DELTA_C;
  float att = DELTA_C / (ld + SMALL_C);

  float* o = out + (size_t)node * 768;
  float sc[3] = {1.f, amp, att};
  #pragma unroll
  for (int si = 0; si < 3; si++) {
    float f = sc[si]; int off = si * 256;
    o[off +   0 + lane] = f * s0;    o[off +  32 + lane] = f * s1;
    o[off +  64 + lane] = f * mean0; o[off +  96 + lane] = f * mean1;
    o[off + 128 + lane] = f * std0;  o[off + 160 + lane] = f * std1;
    o[off + 192 + lane] = f * mx0;   o[off + 224 + lane] = f * mx1;
  }
}

extern "C" void kernel_launch(void* const* d_in, const int* in_sizes, int n_in,
                              void* d_out, int out_size, void* d_ws, size_t ws_size,
                              hipStream_t stream) {
  const float* x      = (const float*)d_in[0];
  const long long* a0 = (const long long*)d_in[1];
  const long long* a1 = (const long long*)d_in[2];
  const long long* a2 = (const long long*)d_in[3];
  const long long* a3 = (const long long*)d_in[4];
  const float* W      = (const float*)d_in[5];
  const float* b      = (const float*)d_in[6];
  float* out          = (float*)d_out;

  // Workspace layout: msgs (400000 x 192 f32), row_start (50001), cnt (50000),
  // cursor (50000), edge_list (400000).
  float* msgs  = (float*)d_ws;
  int* rs      = (int*)(msgs + (size_t)N_EDGES * NCOL);
  int* cnt     = rs + N_NODES + 4;
  int* cursor  = cnt + N_NODES;      // contiguous with cnt -> one zero pass
  int* el      = cursor + N_NODES;

  k_zero <<<(2 * N_NODES + 255) / 256, 256, 0, stream>>>(cnt, 2 * N_NODES);
  k_count<<<(N_EDGES + 255) / 256,     256, 0, stream>>>(a0, a1, a2, a3, cnt);
  k_scan <<<1, 1024, 0, stream>>>(cnt, rs, N_NODES);
  k_fill <<<(N_EDGES + 255) / 256,     256, 0, stream>>>(a0, a1, a2, a3, rs, cursor, el);

  dim3 g(EPT / M_TILE, N_ETYPES);
  k_gemm <<<g, 128, 0, stream>>>(x, a0, a1, a2, a3, W, b, msgs);
  k_aggr <<<(N_NODES + 7) / 8, 256, 0, stream>>>(msgs, rs, el, out);
}